// KML_24300924961295
// MI455X (gfx1250) — compile-verified
//
#include <hip/hip_runtime.h>
#include <math.h>

// MoE-routed 2-layer MLP (D=256 -> H=256 -> D=256, tanh, L2-normalize) on
// gfx1250 using fp32 WMMA (V_WMMA_F32_16X16X4_F32). Samples are bucketed by
// expert so each expert's 512KB of weights is streamed once per 32-sample
// tile (L2-resident across tiles: total weights = 33.5MB << 192MB L2).
// HBM traffic ~42MB -> ~1.8us at 23.3 TB/s; fp32 WMMA keeps full reference
// precision while staying memory-bound.

typedef __attribute__((ext_vector_type(2))) float v2f;
typedef __attribute__((ext_vector_type(8))) float v8f;

#define NUM_E 64
#define DIM   256
#define HID   256
#define TILE_M 32
#define LSTR  (DIM + 4)   // LDS row stride (floats); +4 pad -> conflict-free column reads

// CDNA5 has a hardware V_TANH_F32 (transcendental VALU); use it if the
// toolchain exposes the builtin, else fall back to libm tanhf.
#if __has_builtin(__builtin_amdgcn_tanhf)
#define FAST_TANH(x) __builtin_amdgcn_tanhf(x)
#elif __has_builtin(__builtin_amdgcn_tanh_f32)
#define FAST_TANH(x) __builtin_amdgcn_tanh_f32(x)
#else
#define FAST_TANH(x) tanhf(x)
#endif

// ---------------- bucketing prep kernels ----------------

__global__ void KML_init_kernel(int* counts, int* cursor) {
  int t = threadIdx.x;
  if (t < NUM_E) { counts[t] = 0; cursor[t] = 0; }
}

__global__ void KML_count_kernel(const int* __restrict__ rid, int* counts, int B) {
  int i = blockIdx.x * blockDim.x + threadIdx.x;
  if (i < B) atomicAdd(&counts[rid[i]], 1);
}

__global__ void KML_scan_kernel(const int* __restrict__ counts, int* offsets) {
  if (threadIdx.x == 0) {
    int acc = 0;
    for (int i = 0; i < NUM_E; ++i) { offsets[i] = acc; acc += counts[i]; }
    offsets[NUM_E] = acc;
  }
}

__global__ void KML_scatter_kernel(const int* __restrict__ rid,
                                   const int* __restrict__ offsets,
                                   int* cursor, int* bucket, int B) {
  int i = blockIdx.x * blockDim.x + threadIdx.x;
  if (i < B) {
    int e = rid[i];
    int pos = offsets[e] + atomicAdd(&cursor[e], 1);
    bucket[pos] = i;
  }
}

// ---------------- main fused MoE kernel ----------------

__global__ __launch_bounds__(256, 2) void KML_moe_mlp_kernel(
    const int*   __restrict__ eid,
    const float* __restrict__ emb,
    const float* __restrict__ W1,
    const float* __restrict__ b1,
    const float* __restrict__ W2,
    const float* __restrict__ b2,
    const int*   __restrict__ offsets,
    const int*   __restrict__ bucket,
    float*       __restrict__ out)
{
  __shared__ float smem[TILE_M * LSTR];   // X tile -> H tile -> Y tile (reused)
  __shared__ float norm2[TILE_M];

  const int e     = blockIdx.x;
  const int begin = offsets[e];
  const int count = offsets[e + 1] - begin;
  const int row0  = blockIdx.y * TILE_M;
  if (row0 >= count) return;              // block-uniform exit; EXEC stays full

  const int tid   = threadIdx.x;
  const int lane  = tid & 31;
  const int wave  = tid >> 5;             // 8 waves, each owns 32 output cols
  const int l15   = lane & 15;
  const int half  = lane >> 4;            // K-pair / M-half selector (ISA 7.12.2)
  const int rbase = half * 8;
  const int n0    = wave * 32;

  const float* W1e = W1 + (size_t)e * DIM * HID;
  const float* W2e = W2 + (size_t)e * HID * DIM;

  // Stage X[32 x 256] into LDS (zero rows past count — uniform branch).
  for (int r = 0; r < TILE_M; ++r) {
    int g = row0 + r;
    float v = 0.0f;
    if (g < count) {
      int sid = bucket[begin + g];
      v = emb[(size_t)eid[sid] * DIM + tid];
    }
    smem[r * LSTR + tid] = v;
  }
  // warm L2/L0 for this wave's W2 panel while GEMM1 runs
  __builtin_prefetch(&W2e[n0 + l15], 0, 1);
  __syncthreads();

  // ---- GEMM1: H[32x256] = X @ W1[e] ----
  v8f h00 = {}, h01 = {}, h10 = {}, h11 = {};
  for (int k = 0; k < DIM; k += 4) {
    const int kk = k + 2 * half;
    // A frags: lane = row M, half-wave picks K pair (contiguous -> ds_load_b64)
    v2f a0 = *(const v2f*)&smem[ l15        * LSTR + kk];
    v2f a1 = *(const v2f*)&smem[(l15 + 16)  * LSTR + kk];
    // B frags: lane = col N, half-wave picks K pair (row-major W1 stride HID)
    v2f bA, bB;
    bA.x = W1e[ kk      * HID + (n0 +      l15)];
    bA.y = W1e[(kk + 1) * HID + (n0 +      l15)];
    bB.x = W1e[ kk      * HID + (n0 + 16 + l15)];
    bB.y = W1e[(kk + 1) * HID + (n0 + 16 + l15)];
    h00 = __builtin_amdgcn_wmma_f32_16x16x4_f32(false, a0, false, bA, (short)0, h00, false, false);
    h10 = __builtin_amdgcn_wmma_f32_16x16x4_f32(false, a1, false, bA, (short)0, h10, false, false);
    h01 = __builtin_amdgcn_wmma_f32_16x16x4_f32(false, a0, false, bB, (short)0, h01, false, false);
    h11 = __builtin_amdgcn_wmma_f32_16x16x4_f32(false, a1, false, bB, (short)0, h11, false, false);
  }
  __syncthreads();                        // all waves finished reading X

  // bias + tanh (hardware v_tanh_f32), scatter H back into LDS
  // (C/D layout: lane = N, vgpr index = M within half-wave's M-half)
  {
    float biasA = b1[e * HID + n0 +      l15];
    float biasB = b1[e * HID + n0 + 16 + l15];
#pragma unroll
    for (int r = 0; r < 8; ++r) {
      smem[(     rbase + r) * LSTR + (n0 +      l15)] = FAST_TANH(h00[r] + biasA);
      smem[(16 + rbase + r) * LSTR + (n0 +      l15)] = FAST_TANH(h10[r] + biasA);
      smem[(     rbase + r) * LSTR + (n0 + 16 + l15)] = FAST_TANH(h01[r] + biasB);
      smem[(16 + rbase + r) * LSTR + (n0 + 16 + l15)] = FAST_TANH(h11[r] + biasB);
    }
  }
  __syncthreads();

  // ---- GEMM2: Y[32x256] = H @ W2[e] ----
  v8f y00 = {}, y01 = {}, y10 = {}, y11 = {};
  for (int k = 0; k < HID; k += 4) {
    const int kk = k + 2 * half;
    v2f a0 = *(const v2f*)&smem[ l15       * LSTR + kk];
    v2f a1 = *(const v2f*)&smem[(l15 + 16) * LSTR + kk];
    v2f bA, bB;
    bA.x = W2e[ kk      * DIM + (n0 +      l15)];
    bA.y = W2e[(kk + 1) * DIM + (n0 +      l15)];
    bB.x = W2e[ kk      * DIM + (n0 + 16 + l15)];
    bB.y = W2e[(kk + 1) * DIM + (n0 + 16 + l15)];
    y00 = __builtin_amdgcn_wmma_f32_16x16x4_f32(false, a0, false, bA, (short)0, y00, false, false);
    y10 = __builtin_amdgcn_wmma_f32_16x16x4_f32(false, a1, false, bA, (short)0, y10, false, false);
    y01 = __builtin_amdgcn_wmma_f32_16x16x4_f32(false, a0, false, bB, (short)0, y01, false, false);
    y11 = __builtin_amdgcn_wmma_f32_16x16x4_f32(false, a1, false, bB, (short)0, y11, false, false);
  }
  __syncthreads();                        // all waves finished reading H

  if (tid < TILE_M) norm2[tid] = 0.0f;
  {
    float biasA = b2[e * DIM + n0 +      l15];
    float biasB = b2[e * DIM + n0 + 16 + l15];
#pragma unroll
    for (int r = 0; r < 8; ++r) {
      smem[(     rbase + r) * LSTR + (n0 +      l15)] = y00[r] + biasA;
      smem[(16 + rbase + r) * LSTR + (n0 +      l15)] = y10[r] + biasA;
      smem[(     rbase + r) * LSTR + (n0 + 16 + l15)] = y01[r] + biasB;
      smem[(16 + rbase + r) * LSTR + (n0 + 16 + l15)] = y11[r] + biasB;
    }
  }
  __syncthreads();

  // row-wise sum of squares: 8 threads per row, 32 cols each, ds_add_f32 reduce
  {
    int prow = tid >> 3;
    int pc0  = (tid & 7) * 32;
    float s = 0.0f;
#pragma unroll
    for (int j = 0; j < 32; ++j) {
      float v = smem[prow * LSTR + pc0 + j];
      s += v * v;
    }
    atomicAdd(&norm2[prow], s);
  }
  __syncthreads();

  // normalized, coalesced scatter back to per-sample rows
  for (int r = 0; r < TILE_M; ++r) {
    int g = row0 + r;
    if (g < count) {
      int sid = bucket[begin + g];
      out[(size_t)sid * DIM + tid] = smem[r * LSTR + tid] * rsqrtf(norm2[r]);
    }
  }
}

// ---------------- launcher ----------------

extern "C" void kernel_launch(void* const* d_in, const int* in_sizes, int n_in,
                              void* d_out, int out_size, void* d_ws, size_t ws_size,
                              hipStream_t stream) {
  const int*   eid = (const int*)  d_in[0];   // entity_ids   [B]
  const int*   rid = (const int*)  d_in[1];   // relation_ids [B]
  const float* emb = (const float*)d_in[2];   // [N_ENT, 256]
  const float* W1  = (const float*)d_in[3];   // [64, 256, 256]
  const float* b1  = (const float*)d_in[4];   // [64, 256]
  const float* W2  = (const float*)d_in[5];   // [64, 256, 256]
  const float* b2  = (const float*)d_in[6];   // [64, 256]
  float* out = (float*)d_out;
  const int B = in_sizes[0];

  // workspace layout (ints): counts[64] | offsets[65] | cursor[64] | bucket[B]
  int* counts  = (int*)d_ws;
  int* offsets = counts + NUM_E;
  int* cursor  = offsets + NUM_E + 1;
  int* bucket  = cursor + NUM_E;

  KML_init_kernel   <<<1, 64, 0, stream>>>(counts, cursor);
  KML_count_kernel  <<<(B + 255) / 256, 256, 0, stream>>>(rid, counts, B);
  KML_scan_kernel   <<<1, 1, 0, stream>>>(counts, offsets);
  KML_scatter_kernel<<<(B + 255) / 256, 256, 0, stream>>>(rid, offsets, cursor, bucket, B);

  dim3 grid(NUM_E, (B + TILE_M - 1) / TILE_M);
  KML_moe_mlp_kernel<<<grid, 256, 0, stream>>>(eid, emb, W1, b1, W2, b2,
                                               offsets, bucket, out);
}